// Correction_Module_dense_checksum_22565758173768
// MI455X (gfx1250) — compile-verified
//
#include <hip/hip_runtime.h>

// ---------------------------------------------------------------------------
// ABFT-corrected GEMM verification for MI455X (gfx1250, wave32).
// Fused single pass over C (bandwidth-bound: ~516 MB @ 23.3 TB/s ≈ 22 us).
// CC_check tile computed per-wave with V_WMMA_F32_16X16X4_F32 (f32 precision
// required by rtol=1e-4). C stream uses non-temporal hints + prefetch so L2
// keeps AC/BC (1 MB each) resident for their 256x reuse.
// ---------------------------------------------------------------------------

typedef __attribute__((ext_vector_type(2))) float v2f;
typedef __attribute__((ext_vector_type(4))) float v4f;
typedef __attribute__((ext_vector_type(8))) float v8f;

#define MDIM 8192
#define NDIM 8192
#define DDIM 64
#define WB   4096   // checksum blocks per dim (M/2 = N/2)
#define ATOL 1e-3f
#define RTOL 1e-4f

// ---------------------------------------------------------------------------
// Kernel 1: pairwise row-sum checksums. AC[i,:] = A[2i,:]+A[2i+1,:], same BC.
// One thread per float4 of output: 2 * 4096 * 16 = 131072 threads. Trivial BW.
// ---------------------------------------------------------------------------
__global__ void checksum_kernel(const float* __restrict__ A,
                                const float* __restrict__ B,
                                float* __restrict__ AC,
                                float* __restrict__ BC) {
    int t = blockIdx.x * blockDim.x + threadIdx.x;
    const int half = WB * (DDIM / 4);           // 65536 float4s per matrix
    const float* __restrict__ src = (t < half) ? A : B;
    float* __restrict__ dst       = (t < half) ? AC : BC;
    int i = (t < half) ? t : t - half;
    int row = i >> 4;                            // block row (0..4095)
    int c4  = (i & 15) << 2;                     // float4 column offset
    v4f a = *(const v4f*)(src + (size_t)(2 * row) * DDIM + c4);
    v4f b = *(const v4f*)(src + (size_t)(2 * row + 1) * DDIM + c4);
    *(v4f*)(dst + (size_t)row * DDIM + c4) = a + b;
}

// ---------------------------------------------------------------------------
// Kernel 2: fused verify + correct.
// One wave owns a 32x32 tile of C == a 16x16 tile of 2x2 checksum blocks.
//   Step 1: CC_check tile = BC_tile(16x64) @ AC_tile^T(64x16) via 16 chained
//           V_WMMA_F32_16X16X4_F32 accumulations.
//   Step 2: per accumulator element (lane, vgpr v) -> block (bm, bn):
//           stream the 2x2 C values (two coalesced v2f NT loads), block-sum,
//           compare, and on the rare mismatch recompute B_blk @ A_blk^T.
//   Step 3: coalesced v2f NT stores of the corrected tile.
// ---------------------------------------------------------------------------
__global__ void verify_correct_kernel(const float* __restrict__ A,
                                      const float* __restrict__ B,
                                      const float* __restrict__ C,
                                      float* __restrict__ out,
                                      const float* __restrict__ AC,
                                      const float* __restrict__ BC) {
    const int lane = threadIdx.x & 31;
    const int wave = threadIdx.x >> 5;
    const int tile = blockIdx.x * 8 + wave;      // 65536 tiles total
    const int tile_n = tile & 255;
    const int tile_m = tile >> 8;
    const int bm0 = tile_m * 16;                 // block-row origin
    const int bn0 = tile_n * 16;                 // block-col origin
    const int row0 = tile_m * 32;                // C-row origin
    const int col0 = tile_n * 32;                // C-col origin

    // Prefetch the 32 rows of this C tile (one 128B row segment per lane)
    // while the WMMA chain runs -> global_prefetch_b8.
    __builtin_prefetch(C + (size_t)(row0 + lane) * NDIM + col0, 0, 0);

    // WMMA operand addressing (identical pattern for A- and B-operand):
    //   A-op 16x4 f32: lane<16 holds (M=lane,   K=4s+0..1)
    //                  lane>=16    (M=lane-16, K=4s+2..3)
    //   B-op 4x16 f32: lane<16 holds (N=lane,   K=4s+0..1) etc. (AC^T chunk)
    const int lrow  = lane & 15;
    const int khalf = (lane >> 4) << 1;          // 0 or 2
    const float* __restrict__ bcp = BC + (size_t)(bm0 + lrow) * DDIM + khalf;
    const float* __restrict__ acp = AC + (size_t)(bn0 + lrow) * DDIM + khalf;

    v8f acc = {};
#pragma unroll
    for (int s = 0; s < 16; ++s) {
        v2f a = *(const v2f*)(bcp + 4 * s);
        v2f b = *(const v2f*)(acp + 4 * s);
        // D = A x B + C  (16x16x4 f32), chained accumulation
        acc = __builtin_amdgcn_wmma_f32_16x16x4_f32(
            /*neg_a=*/false, a, /*neg_b=*/false, b,
            /*c_mod=*/(short)0, acc, /*reuse_a=*/false, /*reuse_b=*/false);
    }

    // C/D layout: lane l, VGPR v -> M = v + 8*(l>>4), N = l&15.
    const int bn = bn0 + lrow;
#pragma unroll
    for (int v = 0; v < 8; ++v) {
        const int bm = bm0 + v + ((lane >> 4) << 3);
        const size_t r0 = (size_t)(2 * bm) * NDIM + (size_t)(2 * bn);
        const size_t r1 = r0 + NDIM;
        // Stream 2x2 block: read-once -> non-temporal.
        v2f c0 = __builtin_nontemporal_load((const v2f*)(C + r0));
        v2f c1 = __builtin_nontemporal_load((const v2f*)(C + r1));
        const float check  = acc[v];
        const float actual = (c0.x + c0.y) + (c1.x + c1.y);
        // ~isclose(actual, check, rtol, atol)
        if (fabsf(actual - check) > (ATOL + RTOL * fabsf(check))) {
            // Rare path (~1e-5 density): recompute the 2x2 block of B @ A^T.
            const float* __restrict__ b0 = B + (size_t)(2 * bm) * DDIM;
            const float* __restrict__ b1 = b0 + DDIM;
            const float* __restrict__ a0 = A + (size_t)(2 * bn) * DDIM;
            const float* __restrict__ a1 = a0 + DDIM;
            float s00 = 0.f, s01 = 0.f, s10 = 0.f, s11 = 0.f;
#pragma unroll 4
            for (int d = 0; d < DDIM; d += 4) {
                v4f vb0 = *(const v4f*)(b0 + d);
                v4f vb1 = *(const v4f*)(b1 + d);
                v4f va0 = *(const v4f*)(a0 + d);
                v4f va1 = *(const v4f*)(a1 + d);
                s00 += vb0.x * va0.x + vb0.y * va0.y + vb0.z * va0.z + vb0.w * va0.w;
                s01 += vb0.x * va1.x + vb0.y * va1.y + vb0.z * va1.z + vb0.w * va1.w;
                s10 += vb1.x * va0.x + vb1.y * va0.y + vb1.z * va0.z + vb1.w * va0.w;
                s11 += vb1.x * va1.x + vb1.y * va1.y + vb1.z * va1.z + vb1.w * va1.w;
            }
            c0.x = s00; c0.y = s01;
            c1.x = s10; c1.y = s11;
        }
        __builtin_nontemporal_store(c0, (v2f*)(out + r0));
        __builtin_nontemporal_store(c1, (v2f*)(out + r1));
    }
}

extern "C" void kernel_launch(void* const* d_in, const int* in_sizes, int n_in,
                              void* d_out, int out_size, void* d_ws, size_t ws_size,
                              hipStream_t stream) {
    const float* A = (const float*)d_in[0];   // (8192, 64)
    const float* B = (const float*)d_in[1];   // (8192, 64)
    const float* C = (const float*)d_in[2];   // (8192, 8192)
    float* out = (float*)d_out;               // (8192, 8192)

    // Workspace: AC (1 MB) then BC (1 MB).
    float* AC = (float*)d_ws;
    float* BC = AC + (size_t)WB * DDIM;

    // Phase 1: operand checksums (131072 threads).
    checksum_kernel<<<512, 256, 0, stream>>>(A, B, AC, BC);

    // Phase 2: fused verify+correct. 65536 tiles, 8 waves (tiles) per block.
    verify_correct_kernel<<<8192, 256, 0, stream>>>(A, B, C, out, AC, BC);
}